// SubspaceMethod_24275155157613
// MI455X (gfx1250) — compile-verified
//
#include <hip/hip_runtime.h>
#include <hip/hip_bf16.h>

typedef __attribute__((ext_vector_type(2))) float v2f;
typedef __attribute__((ext_vector_type(8))) float v8f;

#define B_BATCH 4096
#define M_DIM   15
#define T_DIM   200
#define NPAD    16

// ---------------------------------------------------------------------------
// Kernel 1: batched complex covariance via V_WMMA_F32_16X16X4_F32.
// One wave (32 lanes) per batch. A is 16x4 (M x K), B is 4x16 (K x N).
// For the symmetric product X * X^T the A and B register images are identical
// under the ISA's 32-bit operand striping:
//   lane<16 : row = lane,    v0 = K0, v1 = K1
//   lane>=16: row = lane-16, v0 = K2, v1 = K3
// Rr = (Xr Xr^T + Xi Xi^T)/T ; Ri = (Xi Xr^T - Xr Xi^T)/T
// ---------------------------------------------------------------------------
__global__ __launch_bounds__(256) void cov_wmma_kernel(
    const float* __restrict__ xr, const float* __restrict__ xi,
    float* __restrict__ rr, float* __restrict__ ri) {
  const int wave = (blockIdx.x * blockDim.x + threadIdx.x) >> 5;
  const int lane = threadIdx.x & 31;
  const int b = wave;
  if (b >= B_BATCH) return;

  const int half = lane >> 4;   // 0 -> K pair (0,1); 1 -> K pair (2,3)
  const int row  = lane & 15;   // m for A == n for B
  const bool valid = row < M_DIM;

  const float* pr = xr + ((size_t)b * M_DIM + row) * T_DIM;
  const float* pi = xi + ((size_t)b * M_DIM + row) * T_DIM;

  v8f acc_rr = {}; v8f acc_ii = {}; v8f acc_ir = {}; v8f acc_ri = {};

  for (int k0 = 0; k0 < T_DIM; k0 += 4) {
    const int k = k0 + half * 2;
    v2f ar = {0.0f, 0.0f};
    v2f ai = {0.0f, 0.0f};
    if (valid) {
      ar.x = pr[k]; ar.y = pr[k + 1];
      ai.x = pi[k]; ai.y = pi[k + 1];
    }
    acc_rr = __builtin_amdgcn_wmma_f32_16x16x4_f32(false, ar, false, ar, (short)0, acc_rr, false, false);
    acc_ii = __builtin_amdgcn_wmma_f32_16x16x4_f32(false, ai, false, ai, (short)0, acc_ii, false, false);
    acc_ir = __builtin_amdgcn_wmma_f32_16x16x4_f32(false, ai, false, ar, (short)0, acc_ir, false, false);
    acc_ri = __builtin_amdgcn_wmma_f32_16x16x4_f32(false, ar, false, ai, (short)0, acc_ri, false, false);
  }

  // C/D layout: VGPR v, lanes 0-15 -> M=v, lanes 16-31 -> M=v+8; N = lane%16.
  float* rrb = rr + (size_t)b * 256;
  float* rib = ri + (size_t)b * 256;
  const float invT = 1.0f / (float)T_DIM;
#pragma unroll
  for (int v = 0; v < 8; ++v) {
    const int m = v + half * 8;
    const int n = row;
    rrb[m * NPAD + n] = (acc_rr[v] + acc_ii[v]) * invT;
    rib[m * NPAD + n] = (acc_ir[v] - acc_ri[v]) * invT;
  }
}

// ---------------------------------------------------------------------------
// Kernel 2: per-batch complex Hermitian Jacobi eigendecomposition in LDS.
// 8 waves / block, one wave per batch; lanes parallelize over the 15 columns.
// Rotation: factor phase of a_pq (a_pq = |g| e^{i phi}), then real Jacobi
// angle from theta = (a_qq - a_pp) / (2|g|). A' = G^H A G, V' = V G with
//   G = [[c, s], [-e^{-i phi} s, e^{-i phi} c]].
// ---------------------------------------------------------------------------
__global__ __launch_bounds__(256) void eig_jacobi_kernel(
    const float* __restrict__ rr, const float* __restrict__ ri,
    const float* __restrict__ thr_p, const int* __restrict__ ns_p,
    float* __restrict__ out, float* __restrict__ leig_part) {
  __shared__ float Ar[8][M_DIM][NPAD];
  __shared__ float Ai[8][M_DIM][NPAD];
  __shared__ float Vr[8][M_DIM][NPAD];
  __shared__ float Vi[8][M_DIM][NPAD];
  __shared__ float sev[8][NPAD];   // eigenvalues sorted descending

  const int w    = threadIdx.x >> 5;
  const int lane = threadIdx.x & 31;
  const int b    = blockIdx.x * 8 + w;

  const float thr = *thr_p;
  const int   ns  = *ns_p;
  const int   nn  = M_DIM - ns;

  const float* rrb = rr + (size_t)b * 256;
  const float* rib = ri + (size_t)b * 256;

  for (int idx = lane; idx < M_DIM * M_DIM; idx += 32) {
    const int m = idx / M_DIM, n = idx % M_DIM;
    Ar[w][m][n] = rrb[m * NPAD + n];
    Ai[w][m][n] = rib[m * NPAD + n];
    Vr[w][m][n] = (m == n) ? 1.0f : 0.0f;
    Vi[w][m][n] = 0.0f;
  }
  __syncthreads();

  const int j = lane;
  const int NSWEEP = 12;
  for (int sweep = 0; sweep < NSWEEP; ++sweep) {
    for (int p = 0; p < M_DIM - 1; ++p) {
      for (int q = p + 1; q < M_DIM; ++q) {
        // every lane computes the (identical) rotation parameters
        const float gr  = Ar[w][p][q], gi = Ai[w][p][q];
        const float app = Ar[w][p][p], aqq = Ar[w][q][q];
        const float mag = sqrtf(gr * gr + gi * gi);
        float c = 1.0f, s = 0.0f, er = 1.0f, ei = 0.0f;
        if (mag > 1e-12f) {
          er = gr / mag; ei = gi / mag;
          const float theta = (aqq - app) / (2.0f * mag);
          const float t = (theta >= 0.0f ? 1.0f : -1.0f) /
                          (fabsf(theta) + sqrtf(theta * theta + 1.0f));
          c = 1.0f / sqrtf(t * t + 1.0f);
          s = t * c;
        }
        // u = s * conj(e), wv = c * conj(e)
        const float ur = s * er, ui = -s * ei;
        const float wr = c * er, wi = -c * ei;

        if (j < M_DIM) {
          // phase 1: right-multiply columns p,q of A and V by G
          float apr = Ar[w][j][p], api = Ai[w][j][p];
          float aqr = Ar[w][j][q], aqi = Ai[w][j][q];
          Ar[w][j][p] = c * apr - (ur * aqr - ui * aqi);
          Ai[w][j][p] = c * api - (ur * aqi + ui * aqr);
          Ar[w][j][q] = s * apr + (wr * aqr - wi * aqi);
          Ai[w][j][q] = s * api + (wr * aqi + wi * aqr);

          float vpr = Vr[w][j][p], vpi = Vi[w][j][p];
          float vqr = Vr[w][j][q], vqi = Vi[w][j][q];
          Vr[w][j][p] = c * vpr - (ur * vqr - ui * vqi);
          Vi[w][j][p] = c * vpi - (ur * vqi + ui * vqr);
          Vr[w][j][q] = s * vpr + (wr * vqr - wi * vqi);
          Vi[w][j][q] = s * vpi + (wr * vqi + wi * vqr);
        }
        __syncthreads();
        if (j < M_DIM) {
          // phase 2: left-multiply rows p,q by G^H (x = s*e, y = c*e)
          const float xr_ = s * er, xi_ = s * ei;
          const float yr_ = c * er, yi_ = c * ei;
          float apr = Ar[w][p][j], api = Ai[w][p][j];
          float aqr = Ar[w][q][j], aqi = Ai[w][q][j];
          Ar[w][p][j] = c * apr - (xr_ * aqr - xi_ * aqi);
          Ai[w][p][j] = c * api - (xr_ * aqi + xi_ * aqr);
          Ar[w][q][j] = s * apr + (yr_ * aqr - yi_ * aqi);
          Ai[w][q][j] = s * api + (yr_ * aqi + yi_ * aqr);
        }
        __syncthreads();
      }
    }
  }

  // rank eigenvalues descending (rank-by-count, tie-break by index)
  int rank = 0;
  float myev = 0.0f;
  if (j < M_DIM) {
    myev = Ar[w][j][j];
    for (int i = 0; i < M_DIM; ++i) {
      const float e = Ar[w][i][i];
      rank += (e > myev) || (e == myev && i < j);
    }
    sev[w][rank] = myev;
  }
  __syncthreads();

  // output layout (floats, complex interleaved re/im):
  //   signal [B][15][ns][2], noise [B][15][nn][2], source_est int32 [B], l_eig [1]
  const size_t sigBase   = 0;
  const size_t noiseBase = (size_t)B_BATCH * M_DIM * ns * 2;
  const size_t seBase    = noiseBase + (size_t)B_BATCH * M_DIM * nn * 2;

  if (j < M_DIM) {
    if (rank < ns) {
      for (int m = 0; m < M_DIM; ++m) {
        const size_t o = sigBase + (((size_t)b * M_DIM + m) * ns + rank) * 2;
        out[o]     = Vr[w][m][j];
        out[o + 1] = Vi[w][m][j];
      }
    } else {
      const int rk = rank - ns;
      for (int m = 0; m < M_DIM; ++m) {
        const size_t o = noiseBase + (((size_t)b * M_DIM + m) * nn + rk) * 2;
        out[o]     = Vr[w][m][j];
        out[o + 1] = Vi[w][m][j];
      }
    }
  }

  if (lane == 0) {
    const float evmax = sev[w][0];
    int cnt = 0;
    for (int i = 0; i < M_DIM; ++i) cnt += (sev[w][i] / evmax - thr) > 0.0f;
    ((int*)out)[seBase + b] = cnt;
    leig_part[b] = (sev[w][ns - 1] / evmax - thr) * (sev[w][ns] / evmax - thr);
  }
}

// ---------------------------------------------------------------------------
// Kernel 3: deterministic reduction of the 4096 per-batch l_eig partials.
// ---------------------------------------------------------------------------
__global__ __launch_bounds__(256) void leig_reduce_kernel(
    const float* __restrict__ part, const int* __restrict__ ns_p,
    float* __restrict__ out) {
  __shared__ float buf[256];
  const int ns = *ns_p;
  const int nn = M_DIM - ns;
  const size_t leigIdx = (size_t)B_BATCH * M_DIM * ns * 2 +
                         (size_t)B_BATCH * M_DIM * nn * 2 + B_BATCH;
  float s = 0.0f;
  for (int i = threadIdx.x; i < B_BATCH; i += 256) s += part[i];
  buf[threadIdx.x] = s;
  __syncthreads();
  for (int off = 128; off > 0; off >>= 1) {
    if (threadIdx.x < off) buf[threadIdx.x] += buf[threadIdx.x + off];
    __syncthreads();
  }
  if (threadIdx.x == 0) out[leigIdx] = buf[0];
}

// ---------------------------------------------------------------------------
extern "C" void kernel_launch(void* const* d_in, const int* in_sizes, int n_in,
                              void* d_out, int out_size, void* d_ws, size_t ws_size,
                              hipStream_t stream) {
  const float* xr    = (const float*)d_in[0];
  const float* xi    = (const float*)d_in[1];
  const float* thr_p = (const float*)d_in[2];
  const int*   ns_p  = (const int*)d_in[3];
  float* out = (float*)d_out;

  float* rr        = (float*)d_ws;                       // 4096*256 floats
  float* ri        = rr + (size_t)B_BATCH * 256;         // 4096*256 floats
  float* leig_part = ri + (size_t)B_BATCH * 256;         // 4096 floats

  // 1 wave per batch: 4096 waves = 512 blocks of 256 threads (8 waves)
  cov_wmma_kernel<<<B_BATCH / 8, 256, 0, stream>>>(xr, xi, rr, ri);
  eig_jacobi_kernel<<<B_BATCH / 8, 256, 0, stream>>>(rr, ri, thr_p, ns_p, out, leig_part);
  leig_reduce_kernel<<<1, 256, 0, stream>>>(leig_part, ns_p, out);
}